// Generator2_56358560858128
// MI455X (gfx1250) — compile-verified
//
#include <hip/hip_runtime.h>
#include <cstdint>

typedef __attribute__((ext_vector_type(2))) float v2f;
typedef __attribute__((ext_vector_type(8))) float v8f;

#define BN_EPS 1e-3f

// ---------------- utility: zero fill (grid-stride) ----------------
__global__ void zero_f32(float* p, long n) {
    long i = (long)blockIdx.x * blockDim.x + threadIdx.x;
    long stride = (long)gridDim.x * blockDim.x;
    for (; i < n; i += stride) p[i] = 0.0f;
}

// ------------- pack B[K x N] row-major -> Bt[Npad x Kpad], zero padded -------
__global__ void pack_bt(const float* __restrict__ B, float* __restrict__ Bt,
                        int Krows, int Ncols, int Kpad) {
    int k = blockIdx.x * blockDim.x + threadIdx.x;
    int n = blockIdx.y;
    if (k >= Kpad) return;
    float v = (k < Krows && n < Ncols) ? B[(size_t)k * Ncols + n] : 0.0f;
    Bt[(size_t)n * Kpad + k] = v;
}

// ---------------- degree via f32 atomics ----------------
__global__ void deg_kernel(const long long* __restrict__ ei, float* deg, int E) {
    int e = blockIdx.x * blockDim.x + threadIdx.x;
    if (e < E) unsafeAtomicAdd(&deg[(int)ei[(size_t)E + e]], 1.0f);
}

__global__ void dinv_kernel(const float* __restrict__ deg, float* dinv, int n) {
    int i = blockIdx.x * blockDim.x + threadIdx.x;
    if (i < n) dinv[i] = rsqrtf(deg[i] + 1.0f);
}

// ---------------- f32 WMMA GEMM: C[M x Npad] = A[M x Kpad] * Bt^T -------------
// Bt is pre-packed [Npad x Kpad] (transposed, zero padded) -> branch-free loop.
// One wave computes a 16x32 C tile (2 accumulators), A fragment reused 2x.
// Inner loop: three global_load_b64 + two v_wmma per 4 K.
__global__ void __launch_bounds__(128)
gemm_wmma(const float* __restrict__ A, const float* __restrict__ Bt,
          float* __restrict__ C,
          int mtiles, int ngroups, int Kpad, int lda, int ldc) {
    int idx = blockIdx.x * 4 + (threadIdx.x >> 5);
    if (idx >= mtiles * ngroups) return;
    int mt = idx / ngroups, ng = idx % ngroups;
    int lane = threadIdx.x & 31;
    int l = lane & 15, half = lane >> 4;

    const float* Arow = A + (size_t)(mt * 16 + l) * lda;
    const float* B0 = Bt + (size_t)(ng * 32 + l) * Kpad;  // n-tile 0, column l
    const float* B1 = B0 + (size_t)16 * Kpad;             // n-tile 1, column l

    v8f acc0 = {0.f, 0.f, 0.f, 0.f, 0.f, 0.f, 0.f, 0.f};
    v8f acc1 = {0.f, 0.f, 0.f, 0.f, 0.f, 0.f, 0.f, 0.f};
#pragma unroll 2
    for (int k0 = 0; k0 < Kpad; k0 += 4) {
        int ka = k0 + 2 * half;
        v2f a  = *(const v2f*)(Arow + ka);   // A frag: rows l, K = ka, ka+1
        v2f b0 = *(const v2f*)(B0 + ka);     // B frag tile 0
        v2f b1 = *(const v2f*)(B1 + ka);     // B frag tile 1
        acc0 = __builtin_amdgcn_wmma_f32_16x16x4_f32(
            false, a, false, b0, (short)0, acc0, false, false);
        acc1 = __builtin_amdgcn_wmma_f32_16x16x4_f32(
            false, a, false, b1, (short)0, acc1, false, false);
    }

    int rbase = mt * 16 + half * 8;
    int col0 = ng * 32 + l;
#pragma unroll
    for (int v = 0; v < 8; ++v) {
        float* crow = C + (size_t)(rbase + v) * ldc;
        crow[col0]      = acc0[v];
        crow[col0 + 16] = acc1[v];
    }
}

// ---------------- edge aggregation: agg[dst] += h[src] * dinv[src]*dinv[dst] ----
// grid = (E, ceil(Fvalid/256)), block = 256. Edge scalars are uniform per block.
__global__ void agg_kernel(const long long* __restrict__ ei,
                           const float* __restrict__ dinv,
                           const float* __restrict__ h, float* agg,
                           int E, int Fvalid, int Fpad) {
    int e = blockIdx.x;
    int f = blockIdx.y * blockDim.x + threadIdx.x;
    if (f >= Fvalid) return;
    int s = (int)ei[e];
    int d = (int)ei[(size_t)E + e];
    float coef = dinv[s] * dinv[d];
    float val = h[(size_t)s * Fpad + f] * coef;
    unsafeAtomicAdd(&agg[(size_t)d * Fpad + f], val);
}

// ------------- fused epilogue: x = sigmoid(BN(relu(agg + h*dinv^2 + b))) -------
__global__ void act_kernel(float* x, const float* __restrict__ h,
                           const float* __restrict__ dinv,
                           const float* __restrict__ b, const float* __restrict__ g,
                           const float* __restrict__ beta, const float* __restrict__ rm,
                           const float* __restrict__ rv,
                           int N, int Fvalid, int Fpad) {
    long tid = (long)blockIdx.x * blockDim.x + threadIdx.x;
    long total = (long)N * Fpad;
    if (tid >= total) return;
    int i = (int)(tid / Fpad);
    int f = (int)(tid % Fpad);
    float out = 0.0f;
    if (f < Fvalid) {
        float di = dinv[i];
        float v = x[tid] + h[tid] * di * di + b[f];
        v = fmaxf(v, 0.0f);
        v = (v - rm[f]) * rsqrtf(rv[f] + BN_EPS) * g[f] + beta[f];
        out = 1.0f / (1.0f + __expf(-v));
    }
    x[tid] = out;  // pads set to 0 so later GEMM/gram K-loops need no masking
}

// ---------------- gram: out = X^T X via f32 WMMA, K split + atomics ----------
// grid = (ntiles*ntiles, kchunks), block = 32 (one wave).
__global__ void __launch_bounds__(32)
gram_wmma(const float* __restrict__ X, float* out,
          int ntiles, int ld, int kchunk, int Nvalid) {
    int tile = blockIdx.x;
    int it = tile / ntiles, jt = tile % ntiles;
    int k0base = blockIdx.y * kchunk;
    int lane = threadIdx.x & 31;
    int l = lane & 15, half = lane >> 4;
    int icol = it * 16 + l;
    int jcol = jt * 16 + l;

    v8f acc = {0.f, 0.f, 0.f, 0.f, 0.f, 0.f, 0.f, 0.f};
#pragma unroll 2
    for (int kk = 0; kk < kchunk; kk += 4) {
        int ka = k0base + kk + 2 * half;
        const float* r0 = X + (size_t)ka * ld;
        const float* r1 = r0 + ld;
        v2f a; a.x = r0[icol]; a.y = r1[icol];   // A = X^T tile (16x4)
        v2f b; b.x = r0[jcol]; b.y = r1[jcol];   // B = X tile  (4x16)
        acc = __builtin_amdgcn_wmma_f32_16x16x4_f32(
            false, a, false, b, (short)0, acc, false, false);
    }
#pragma unroll
    for (int v = 0; v < 8; ++v) {
        int row = it * 16 + half * 8 + v;
        int c   = jt * 16 + l;
        if (row < Nvalid && c < Nvalid)
            unsafeAtomicAdd(&out[row * Nvalid + c], acc[v]);
    }
}

extern "C" void kernel_launch(void* const* d_in, const int* in_sizes, int n_in,
                              void* d_out, int out_size, void* d_ws, size_t ws_size,
                              hipStream_t stream) {
    const float*     x     = (const float*)d_in[0];
    const long long* ei    = (const long long*)d_in[1];
    const float*     W1    = (const float*)d_in[2];
    const float*     b1    = (const float*)d_in[3];
    const float*     g1    = (const float*)d_in[4];
    const float*     beta1 = (const float*)d_in[5];
    const float*     rm1   = (const float*)d_in[6];
    const float*     rv1   = (const float*)d_in[7];
    const float*     W2    = (const float*)d_in[8];
    const float*     b2    = (const float*)d_in[9];
    const float*     g2    = (const float*)d_in[10];
    const float*     beta2 = (const float*)d_in[11];
    const float*     rm2   = (const float*)d_in[12];
    const float*     rv2   = (const float*)d_in[13];
    float* out = (float*)d_out;

    const int N = 50000, E = 800000;
    const int K1 = 160;
    const int F1 = 536, F1p = 544;   // layer-1 width padded: 34 N-tiles (17 pairs)
    const int F2 = 268, F2p = 288;   // layer-2 width padded: 18 N-tiles (9 pairs)

    float* ws   = (float*)d_ws;
    float* hbuf = ws;                               // N * F1p
    float* xbuf = hbuf + (size_t)N * F1p;           // N * F1p
    float* dinv = xbuf + (size_t)N * F1p;           // N
    float* deg  = dinv + N;                         // N
    float* W1t  = deg + N;                          // F1p * K1
    float* W2t  = W1t + (size_t)F1p * K1;           // F2p * F1p
    float* h2   = hbuf;                             // reuse: N * F2p
    float* agg2 = xbuf;                             // reuse: N * F2p (x2 in place)

    // ---- init (graph-replay safe: re-zero every call) ----
    zero_f32<<<2048, 256, 0, stream>>>(deg, N);
    zero_f32<<<8192, 256, 0, stream>>>(xbuf, (long)N * F1p);
    zero_f32<<<512,  256, 0, stream>>>(out, (long)F2 * F2);

    // ---- pack transposed, zero-padded weights for branch-free WMMA B loads ----
    pack_bt<<<dim3((K1 + 255) / 256, F1p), 256, 0, stream>>>(W1, W1t, K1, F1, K1);
    pack_bt<<<dim3((F1p + 255) / 256, F2p), 256, 0, stream>>>(W2, W2t, F1, F2, F1p);

    // ---- degree / normalization ----
    deg_kernel<<<(E + 255) / 256, 256, 0, stream>>>(ei, deg, E);
    dinv_kernel<<<(N + 255) / 256, 256, 0, stream>>>(deg, dinv, N);

    // ---- layer 1: h1 = x @ W1 ----
    const int mt = N / 16;            // 3125 (exact)
    const int ng1 = F1p / 32;         // 17 groups of 2 N-tiles
    gemm_wmma<<<(mt * ng1 + 3) / 4, 128, 0, stream>>>(
        x, W1t, hbuf, mt, ng1, K1, K1, F1p);

    agg_kernel<<<dim3(E, (F1 + 255) / 256), 256, 0, stream>>>(
        ei, dinv, hbuf, xbuf, E, F1, F1p);

    act_kernel<<<(int)(((long)N * F1p + 255) / 256), 256, 0, stream>>>(
        xbuf, hbuf, dinv, b1, g1, beta1, rm1, rv1, N, F1, F1p);

    // ---- layer 2: h2 = x1 @ W2 (K padded to 544; x1 pad cols are zero) ----
    const int ng2 = F2p / 32;         // 9 groups of 2 N-tiles
    gemm_wmma<<<(mt * ng2 + 3) / 4, 128, 0, stream>>>(
        xbuf, W2t, h2, mt, ng2, F1p, F1p, F2p);

    // x1 fully consumed by GEMM-2 -> reuse xbuf for agg2/x2
    zero_f32<<<8192, 256, 0, stream>>>(agg2, (long)N * F2p);
    agg_kernel<<<dim3(E, (F2 + 255) / 256), 256, 0, stream>>>(
        ei, dinv, h2, agg2, E, F2, F2p);

    act_kernel<<<(int)(((long)N * F2p + 255) / 256), 256, 0, stream>>>(
        agg2, h2, dinv, b2, g2, beta2, rm2, rv2, N, F2, F2p);

    // ---- gram: out = x2^T x2, K=50000 split into 100 chunks of 500 ----
    gram_wmma<<<dim3(17 * 17, 100), 32, 0, stream>>>(
        agg2, out, 17, F2p, 500, F2);
}